// MLPMixer_1623497638486
// MI455X (gfx1250) — compile-verified
//
#include <hip/hip_runtime.h>
#include <cstdint>
#include <cstddef>

// ---------------------------------------------------------------------------
// CDNA5 (gfx1250) MLP-Mixer block, fp32 path via V_WMMA_F32_16X16X4_F32.
// ---------------------------------------------------------------------------

typedef float v2f __attribute__((ext_vector_type(2)));
typedef float v8f __attribute__((ext_vector_type(8)));
// align(4) lets the compiler use wide loads when provably aligned, split otherwise
typedef float f4u __attribute__((ext_vector_type(4), aligned(4)));

#define TILE_BM 128
#define TILE_BN 128
#define TILE_BK 16
#define LDS_PAD 8   // row stride 136 floats = 544B: 16B-aligned rows, disjoint bank groups for the two half-waves

__device__ __forceinline__ float gelu_exact(float v) {
    return 0.5f * v * (1.0f + erff(v * 0.70710678118654752f));
}

// ---------------------------------------------------------------------------
// Conditional LayerNorm: one 256-thread block per (b, s) row of length D.
// gamma = cln[task][0:D], beta = cln[task][D:2D]
// ---------------------------------------------------------------------------
__global__ __launch_bounds__(256)
void ln_kernel(const float* __restrict__ x, const float* __restrict__ cln,
               const int* __restrict__ tasks, float* __restrict__ out,
               int S, int D)
{
    __shared__ float r1[256];
    __shared__ float r2[256];

    const int row  = blockIdx.x;
    const int b    = row / S;
    const int task = tasks[b];
    const float* __restrict__ xp = x + (size_t)row * D;
    float* __restrict__ op = out + (size_t)row * D;
    const float* __restrict__ gam = cln + (size_t)task * 2 * D;

    float s = 0.0f, ss = 0.0f;
    for (int i = threadIdx.x; i < D; i += 256) {
        float v = xp[i];
        s += v;
        ss += v * v;
    }
    r1[threadIdx.x] = s;
    r2[threadIdx.x] = ss;
    __syncthreads();
    #pragma unroll
    for (int off = 128; off >= 1; off >>= 1) {
        if (threadIdx.x < off) {
            r1[threadIdx.x] += r1[threadIdx.x + off];
            r2[threadIdx.x] += r2[threadIdx.x + off];
        }
        __syncthreads();
    }
    const float invD = 1.0f / (float)D;
    const float mu   = r1[0] * invD;
    const float var  = r2[0] * invD - mu * mu;
    const float rstd = rsqrtf(var + 1e-5f);

    for (int i = threadIdx.x; i < D; i += 256) {
        float v = (xp[i] - mu) * rstd;
        op[i] = gam[i] * v + gam[D + i];
    }
}

// ---------------------------------------------------------------------------
// Tiled WMMA GEMM (fp32, 16x16x4):
//   C[b] = op( A[b] (MxK) * B[b] (KxN or, if BT, B is NxK row-major read as K x N) )
//   optional exact GELU, optional residual add.
// Block tile 128x128, BK=16. 8 waves, each 32(M) x 64(N) = 2x4 v8f accumulators.
// Per-operand batch indexing: ptr + task*taskStride + b*batchStride.
// ---------------------------------------------------------------------------
template<bool BT, bool DO_GELU, bool DO_RES>
__global__ __launch_bounds__(256)
void gemm_wmma_f32(const float* __restrict__ Abase, size_t aTask, size_t aBatch,
                   const float* __restrict__ Bbase, size_t bTask, size_t bBatch,
                   float* __restrict__ Cbase, size_t cBatch,
                   const float* __restrict__ Rbase, size_t rBatch,
                   const int* __restrict__ tasks,
                   int M, int N, int K, int lda, int ldb, int ldc)
{
    __shared__ __align__(16) float As[TILE_BK][TILE_BM + LDS_PAD];
    __shared__ __align__(16) float Bs[TILE_BK][TILE_BN + LDS_PAD];

    const int b    = blockIdx.z;
    const int task = tasks[b];
    const float* __restrict__ Ap = Abase + (size_t)task * aTask + (size_t)b * aBatch;
    const float* __restrict__ Bp = Bbase + (size_t)task * bTask + (size_t)b * bBatch;
    float* __restrict__ Cp = Cbase + (size_t)b * cBatch;
    const float* __restrict__ Rp = DO_RES ? (Rbase + (size_t)b * rBatch) : nullptr;

    const int blockM = blockIdx.y * TILE_BM;
    const int blockN = blockIdx.x * TILE_BN;

    const int t    = threadIdx.x;
    const int lane = t & 31;
    const int wave = t >> 5;
    const int waveM = (wave >> 1) * 32;  // 4 waves along M
    const int waveN = (wave & 1) * 64;   // 2 waves along N
    const int half = lane >> 4;          // 0: lanes 0-15 (K+0/1), 1: lanes 16-31 (K+2/3)
    const int l15  = lane & 15;

    v8f acc[2][4] = {};

    for (int k0 = 0; k0 < K; k0 += TILE_BK) {
        // ---- A tile: A is MxK row-major; transpose-store into As[k][m] ----
        #pragma unroll
        for (int it = 0; it < 2; ++it) {
            int c  = t + it * 256;      // 512 chunks of 4 K-values
            int m  = c >> 2;            // 0..127
            int kg = (c & 3) << 2;      // 0,4,8,12
            int gm = blockM + m;
            int gk = k0 + kg;
            float vx = 0.f, vy = 0.f, vz = 0.f, vw = 0.f;
            if (gm < M) {
                const float* p = Ap + (size_t)gm * lda;
                if (gk + 3 < K) {
                    f4u v = *(const f4u*)(p + gk);
                    vx = v.x; vy = v.y; vz = v.z; vw = v.w;
                } else {
                    if (gk + 0 < K) vx = p[gk + 0];
                    if (gk + 1 < K) vy = p[gk + 1];
                    if (gk + 2 < K) vz = p[gk + 2];
                    if (gk + 3 < K) vw = p[gk + 3];
                }
            }
            As[kg + 0][m] = vx;
            As[kg + 1][m] = vy;
            As[kg + 2][m] = vz;
            As[kg + 3][m] = vw;
        }

        // ---- B tile ----
        if constexpr (!BT) {
            // B is KxN row-major: Bs[k][n] = B[k0+k][blockN+n]
            #pragma unroll
            for (int it = 0; it < 2; ++it) {
                int c  = t + it * 256;
                int k  = c >> 5;          // 0..15
                int ng = (c & 31) << 2;   // 0..124
                int gk = k0 + k;
                int gn = blockN + ng;
                float vx = 0.f, vy = 0.f, vz = 0.f, vw = 0.f;
                if (gk < K) {
                    const float* p = Bp + (size_t)gk * ldb;
                    if (gn + 3 < N) {
                        f4u v = *(const f4u*)(p + gn);
                        vx = v.x; vy = v.y; vz = v.z; vw = v.w;
                    } else {
                        if (gn + 0 < N) vx = p[gn + 0];
                        if (gn + 1 < N) vy = p[gn + 1];
                        if (gn + 2 < N) vz = p[gn + 2];
                        if (gn + 3 < N) vw = p[gn + 3];
                    }
                }
                *(float4*)(&Bs[k][ng]) = make_float4(vx, vy, vz, vw);  // 16B-aligned LDS store
            }
        } else {
            // B is NxK row-major (NT): Bs[k][n] = Bt[blockN+n][k0+k]
            #pragma unroll
            for (int it = 0; it < 2; ++it) {
                int c  = t + it * 256;
                int n  = c >> 2;          // 0..127
                int kg = (c & 3) << 2;
                int gn = blockN + n;
                int gk = k0 + kg;
                float vx = 0.f, vy = 0.f, vz = 0.f, vw = 0.f;
                if (gn < N) {
                    const float* p = Bp + (size_t)gn * ldb;
                    if (gk + 3 < K) {
                        f4u v = *(const f4u*)(p + gk);
                        vx = v.x; vy = v.y; vz = v.z; vw = v.w;
                    } else {
                        if (gk + 0 < K) vx = p[gk + 0];
                        if (gk + 1 < K) vy = p[gk + 1];
                        if (gk + 2 < K) vz = p[gk + 2];
                        if (gk + 3 < K) vw = p[gk + 3];
                    }
                }
                Bs[kg + 0][n] = vx;
                Bs[kg + 1][n] = vy;
                Bs[kg + 2][n] = vz;
                Bs[kg + 3][n] = vw;
            }
        }
        __syncthreads();

        // ---- compute: 4 K-steps of 4, 2x4 WMMA tiles each ----
        #pragma unroll
        for (int kk = 0; kk < TILE_BK; kk += 4) {
            const int kA = kk + 2 * half;   // ISA A/B layout: half-wave owns K pairs {0,1}/{2,3}
            v2f afrag[2], bfrag[4];
            #pragma unroll
            for (int mi = 0; mi < 2; ++mi) {
                int m = waveM + 16 * mi + l15;
                afrag[mi].x = As[kA][m];
                afrag[mi].y = As[kA + 1][m];
            }
            #pragma unroll
            for (int ni = 0; ni < 4; ++ni) {
                int n = waveN + 16 * ni + l15;
                bfrag[ni].x = Bs[kA][n];
                bfrag[ni].y = Bs[kA + 1][n];
            }
            #pragma unroll
            for (int mi = 0; mi < 2; ++mi) {
                #pragma unroll
                for (int ni = 0; ni < 4; ++ni) {
                    acc[mi][ni] = __builtin_amdgcn_wmma_f32_16x16x4_f32(
                        false, afrag[mi], false, bfrag[ni],
                        (short)0, acc[mi][ni], false, false);
                }
            }
        }
        __syncthreads();
    }

    // ---- epilogue: C layout per ISA (VGPR i -> M = i + 8*half, N = lane&15) ----
    #pragma unroll
    for (int mi = 0; mi < 2; ++mi) {
        const int mBase = blockM + waveM + 16 * mi + 8 * half;
        #pragma unroll
        for (int ni = 0; ni < 4; ++ni) {
            const int col = blockN + waveN + 16 * ni + l15;
            if (col < N) {
                #pragma unroll
                for (int i = 0; i < 8; ++i) {
                    const int row = mBase + i;
                    if (row < M) {
                        float v = acc[mi][ni][i];
                        if constexpr (DO_GELU) v = gelu_exact(v);
                        const size_t idx = (size_t)row * ldc + col;
                        if constexpr (DO_RES) v += Rp[idx];
                        Cp[idx] = v;
                    }
                }
            }
        }
    }
}

// ---------------------------------------------------------------------------
// Host launcher.
// Inputs (setup_inputs order): x, tasks_id, cln1, cln2, w1_tok, w2_tok, w1_ch, w2_ch
// ---------------------------------------------------------------------------
extern "C" void kernel_launch(void* const* d_in, const int* in_sizes, int n_in,
                              void* d_out, int out_size, void* d_ws, size_t ws_size,
                              hipStream_t stream)
{
    const float* x      = (const float*)d_in[0];
    const int*   tid    = (const int*)  d_in[1];
    const float* cln1   = (const float*)d_in[2];
    const float* cln2   = (const float*)d_in[3];
    const float* w1_tok = (const float*)d_in[4];
    const float* w2_tok = (const float*)d_in[5];
    const float* w1_ch  = (const float*)d_in[6];
    const float* w2_ch  = (const float*)d_in[7];
    float* out = (float*)d_out;

    constexpr int Bb = 64, S = 1024, D = 768;
    constexpr int IS = 307, ID = 230;   // int(0.3*S), int(0.3*D)

    // workspace: h = LN output (B,S,D); g = GELU intermediate (max(B*IS*D, B*S*ID))
    float* h = (float*)d_ws;
    float* g = h + (size_t)Bb * S * D;

    const dim3 blk(256);

    // 1) h = cond_ln(x, cln1)
    ln_kernel<<<Bb * S, blk, 0, stream>>>(x, cln1, tid, h, S, D);

    // 2) g = gelu( w1_tok[task] (IS x S) * h (S x D) )        M=IS N=D K=S (NN)
    gemm_wmma_f32<false, true, false>
        <<<dim3((D + TILE_BN - 1) / TILE_BN, (IS + TILE_BM - 1) / TILE_BM, Bb), blk, 0, stream>>>(
        w1_tok, (size_t)IS * S, 0,
        h,      0,              (size_t)S * D,
        g,      (size_t)IS * D,
        nullptr, 0, tid,
        IS, D, S, S, D, D);

    // 3) out = x + w2_tok[task] (S x IS) * g (IS x D)         M=S N=D K=IS (NN + residual)
    gemm_wmma_f32<false, false, true>
        <<<dim3((D + TILE_BN - 1) / TILE_BN, (S + TILE_BM - 1) / TILE_BM, Bb), blk, 0, stream>>>(
        w2_tok, (size_t)S * IS, 0,
        g,      0,              (size_t)IS * D,
        out,    (size_t)S * D,
        x,      (size_t)S * D, tid,
        S, D, IS, IS, D, D);

    // 4) h = cond_ln(out, cln2)
    ln_kernel<<<Bb * S, blk, 0, stream>>>(out, cln2, tid, h, S, D);

    // 5) g = gelu( h (S x D) * w1_ch[task]^T (D x ID) )       M=S N=ID K=D (NT)
    gemm_wmma_f32<true, true, false>
        <<<dim3((ID + TILE_BN - 1) / TILE_BN, (S + TILE_BM - 1) / TILE_BM, Bb), blk, 0, stream>>>(
        h,     0,              (size_t)S * D,
        w1_ch, (size_t)ID * D, 0,
        g,     (size_t)S * ID,
        nullptr, 0, tid,
        S, ID, D, D, D, ID);

    // 6) out = out + g (S x ID) * w2_ch[task]^T (ID x D)      M=S N=D K=ID (NT + residual, in-place)
    gemm_wmma_f32<true, false, true>
        <<<dim3((D + TILE_BN - 1) / TILE_BN, (S + TILE_BM - 1) / TILE_BM, Bb), blk, 0, stream>>>(
        g,     0,              (size_t)S * ID,
        w2_ch, (size_t)D * ID, 0,
        out,   (size_t)S * D,
        out,   (size_t)S * D, tid,
        S, D, ID, ID, ID, D);
}